// yolov1_loss_30279519437582
// MI455X (gfx1250) — compile-verified
//
#include <hip/hip_runtime.h>
#include <stdint.h>

// YOLOv1 loss on MI455X (gfx1250): pure HBM-bound streaming reduction.
// 192.6 MB in -> 5 floats out; floor ~8.3us @ 23.3 TB/s.
// Kernel 1: async global->LDS staging (ASYNCcnt path), per-cell math from LDS,
//           deterministic tree reduction -> 5 partials per block.
// Kernel 2: per-channel strided accumulation + V_WMMA_F32_16X16X4_F32 with
//           A=ones used as a 64-element cross-lane reduction engine.

#define NCELLS          802816      // 4096 * 14 * 14
#define CELLS_PER_BLOCK 128
#define NBLOCKS1        (NCELLS / CELLS_PER_BLOCK)   // 6272 = 196 * 32
#define BLK1            128
#define BLK2            256
#define NCLS            20

typedef __attribute__((ext_vector_type(2))) float v2f;
typedef __attribute__((ext_vector_type(8))) float v8f;

__device__ __forceinline__ uint32_t lds_off_u32(const void* p) {
  // Flat LDS pointers carry the wave-relative LDS byte offset in the low 32 bits.
  return (uint32_t)(uintptr_t)p;
}

__device__ __forceinline__ void async_b128(uint32_t lds, uint64_t gaddr) {
  asm volatile("global_load_async_to_lds_b128 %0, %1, off"
               :: "v"(lds), "v"(gaddr)
               : "memory");
}

template <int NBYTES>
__device__ __forceinline__ void stage_stream(uint32_t lds_base, const float* g, int tid) {
  uint64_t gb = (uint64_t)(uintptr_t)g;
#pragma unroll
  for (int o = 0; o < NBYTES; o += BLK1 * 16) {
    int off = o + tid * 16;
    if ((NBYTES % (BLK1 * 16)) == 0 || off < NBYTES) {
      async_b128(lds_base + (uint32_t)off, gb + (uint64_t)off);
    }
  }
}

__device__ __forceinline__ float iou_pair(float4 t, float4 p) {
  // corners: xy +/- 0.5*wh^2  (matches _to_corners in the reference)
  float thw = 0.5f * t.z * t.z, thh = 0.5f * t.w * t.w;
  float phw = 0.5f * p.z * p.z, phh = 0.5f * p.w * p.w;
  float tx0 = t.x - thw, ty0 = t.y - thh, tx1 = t.x + thw, ty1 = t.y + thh;
  float px0 = p.x - phw, py0 = p.y - phh, px1 = p.x + phw, py1 = p.y + phh;
  float ltx = fmaxf(tx0, px0), lty = fmaxf(ty0, py0);
  float rbx = fminf(tx1, px1), rby = fminf(ty1, py1);
  float wx = fmaxf(rbx - ltx, 0.0f), wy = fmaxf(rby - lty, 0.0f);
  float inter = wx * wy;
  float a1 = (tx1 - tx0) * (ty1 - ty0);
  float a2 = (px1 - px0) * (py1 - py0);
  return inter / (a1 + a2 - inter);
}

__global__ __launch_bounds__(BLK1) void yolo_partial_kernel(
    const float* __restrict__ pred_cls,  const float* __restrict__ pred_resp,
    const float* __restrict__ pred_bb,   const float* __restrict__ label_cls,
    const float* __restrict__ label_resp,const float* __restrict__ label_bb,
    float* __restrict__ ws)
{
  __shared__ float s_pcls[CELLS_PER_BLOCK * NCLS];  // 10240 B
  __shared__ float s_lcls[CELLS_PER_BLOCK * NCLS];  // 10240 B
  __shared__ float s_pbb [CELLS_PER_BLOCK * 8];     //  4096 B
  __shared__ float s_lbb [CELLS_PER_BLOCK * 8];     //  4096 B
  __shared__ float s_pr  [CELLS_PER_BLOCK * 2];     //  1024 B
  __shared__ float s_lr  [CELLS_PER_BLOCK * 2];     //  1024 B
  __shared__ float s_red[4][5];

  const int    tid   = threadIdx.x;
  const size_t cell0 = (size_t)blockIdx.x * CELLS_PER_BLOCK;

  // --- async stage: global -> LDS (tracked on ASYNCcnt, no dest VGPRs held)
  stage_stream<CELLS_PER_BLOCK * NCLS * 4>(lds_off_u32(s_pcls), pred_cls  + cell0 * NCLS, tid);
  stage_stream<CELLS_PER_BLOCK * NCLS * 4>(lds_off_u32(s_lcls), label_cls + cell0 * NCLS, tid);
  stage_stream<CELLS_PER_BLOCK * 8    * 4>(lds_off_u32(s_pbb),  pred_bb   + cell0 * 8,    tid);
  stage_stream<CELLS_PER_BLOCK * 8    * 4>(lds_off_u32(s_lbb),  label_bb  + cell0 * 8,    tid);
  stage_stream<CELLS_PER_BLOCK * 2    * 4>(lds_off_u32(s_pr),   pred_resp + cell0 * 2,    tid);
  stage_stream<CELLS_PER_BLOCK * 2    * 4>(lds_off_u32(s_lr),   label_resp+ cell0 * 2,    tid);

  asm volatile("s_wait_asynccnt 0x0" ::: "memory");  // my wave's copies done
  __syncthreads();                                   // everyone's copies done

  // --- per-cell math (thread tid <-> cell tid), all loads from LDS, 16B aligned
  float pr0 = s_pr[tid * 2 + 0], pr1 = s_pr[tid * 2 + 1];
  float lr0 = s_lr[tid * 2 + 0], lr1 = s_lr[tid * 2 + 1];

  float obj   = (lr0 > 0.0f) ? 1.0f : 0.0f;
  float noobj = (lr0 < 1.0f) ? 1.0f : 0.0f;

  // class loss: sum over 20 classes of (pc - lc)^2
  const float4* pc4 = (const float4*)(s_pcls + tid * NCLS);
  const float4* lc4 = (const float4*)(s_lcls + tid * NCLS);
  float cls = 0.0f;
#pragma unroll
  for (int q = 0; q < 5; ++q) {
    float4 a = pc4[q], b = lc4[q];
    float d0 = a.x - b.x, d1 = a.y - b.y, d2 = a.z - b.z, d3 = a.w - b.w;
    cls += d0 * d0 + d1 * d1 + d2 * d2 + d3 * d3;
  }

  // boxes
  const float4* pb4 = (const float4*)(s_pbb + tid * 8);
  const float4* lb4 = (const float4*)(s_lbb + tid * 8);
  float4 pb0 = pb4[0], pb1 = pb4[1];
  float4 tb0 = lb4[0], tb1 = lb4[1];

  float iou0 = iou_pair(tb0, pb0);
  float iou1 = iou_pair(tb1, pb1);
  bool  b1   = iou1 > iou0;              // jnp.argmax: first max wins ties -> box 0
  float miou = fmaxf(iou0, iou1);

  float4 bp  = b1 ? pb1 : pb0;
  float4 bt  = b1 ? tb1 : tb0;
  float  prb = b1 ? pr1 : pr0;
  float  lrb = b1 ? lr1 : lr0;

  float c0 = bp.x - bt.x, c1 = bp.y - bt.y, c2 = bp.z - bt.z, c3 = bp.w - bt.w;
  float e0 = pr0 - lr0,  e1 = pr1 - lr1;
  float dp = prb - miou;
  float di = miou - lrb;

  float acc[5];
  acc[0] = obj   * (c0 * c0 + c1 * c1 + c2 * c2 + c3 * c3);  // coord
  acc[1] = obj   * cls;                                      // cls
  acc[2] = obj   * dp * dp;                                  // pobj
  acc[3] = noobj * (e0 * e0 + e1 * e1);                      // nobj
  acc[4] = obj   * di * di;                                  // iou

  // --- wave32 tree reduce, then cross-wave via LDS (deterministic order)
#pragma unroll
  for (int j = 0; j < 5; ++j) {
#pragma unroll
    for (int off = 16; off > 0; off >>= 1) {
      acc[j] += __shfl_down(acc[j], off, 32);
    }
  }
  const int wv = tid >> 5, ln = tid & 31;
  if (ln == 0) {
#pragma unroll
    for (int j = 0; j < 5; ++j) s_red[wv][j] = acc[j];
  }
  __syncthreads();
  if (tid < 5) {
    float r = s_red[0][tid] + s_red[1][tid] + s_red[2][tid] + s_red[3][tid];
    ws[(size_t)blockIdx.x * 5 + tid] = r;
  }
}

__global__ __launch_bounds__(BLK2) void yolo_final_kernel(
    const float* __restrict__ ws, float* __restrict__ out)
{
  __shared__ float s_fin[5];
  const int tid = threadIdx.x;
  const int wv  = tid >> 5, ln = tid & 31;

  if (wv < 5) {
    const int j = wv;  // one loss channel per wave

    // strided per-lane accumulation: 6272 partials -> 32 lane values
    float p = 0.0f;
    for (int t = 0; t < NBLOCKS1 / 32; ++t) {       // 196 iterations, no tail
      p += ws[(size_t)(ln + 32 * t) * 5 + j];
    }

#if __has_builtin(__builtin_amdgcn_wmma_f32_16x16x4_f32)
    // Cross-lane reduce via the matrix unit: D = ones(16x4) x B(4x16) + 0
    // => D[m,n] = sum_k B[k,n]; sum_n D[0,n] = sum of all 64 B elements.
    // B VGPR0 = lane partial, VGPR1 = 0: every lane value lands at a unique
    // (k,n) slot, so the grand total is exact regardless of the lane->(k,n) map.
    v2f amat; amat[0] = 1.0f; amat[1] = 1.0f;
    v2f bmat; bmat[0] = p;    bmat[1] = 0.0f;
    v8f dacc = {};
    dacc = __builtin_amdgcn_wmma_f32_16x16x4_f32(
        /*neg_a=*/false, amat, /*neg_b=*/false, bmat,
        /*c_mod=*/(short)0, dacc, /*reuse_a=*/false, /*reuse_b=*/false);
    // acc[0]: lanes 0..15 hold D[0, n=lane] = the 16 column sums
    float v = dacc[0];
    v += __shfl_down(v, 8, 32);
    v += __shfl_down(v, 4, 32);
    v += __shfl_down(v, 2, 32);
    v += __shfl_down(v, 1, 32);
#else
    float v = p;
    for (int off = 16; off > 0; off >>= 1) v += __shfl_down(v, off, 32);
#endif
    if (ln == 0) s_fin[j] = v;
  }
  __syncthreads();

  if (tid == 0) {
    const float inv_bs = 1.0f / 4096.0f;
    out[0] = 5.0f * s_fin[0] * inv_bs;   // L_COORD * coord_loss
    out[1] = s_fin[1] * inv_bs;          // cls_loss
    out[2] = s_fin[2] * inv_bs;          // pobj_loss
    out[3] = 0.5f * s_fin[3] * inv_bs;   // L_NOOBJ * nobj_loss
    out[4] = s_fin[4] * inv_bs;          // iou_loss
  }
}

extern "C" void kernel_launch(void* const* d_in, const int* in_sizes, int n_in,
                              void* d_out, int out_size, void* d_ws, size_t ws_size,
                              hipStream_t stream) {
  (void)in_sizes; (void)n_in; (void)out_size; (void)ws_size;
  const float* pred_cls   = (const float*)d_in[0];
  const float* pred_resp  = (const float*)d_in[1];
  const float* pred_bb    = (const float*)d_in[2];
  const float* label_cls  = (const float*)d_in[3];
  const float* label_resp = (const float*)d_in[4];
  const float* label_bb   = (const float*)d_in[5];
  float* ws  = (float*)d_ws;
  float* out = (float*)d_out;

  hipLaunchKernelGGL(yolo_partial_kernel, dim3(NBLOCKS1), dim3(BLK1), 0, stream,
                     pred_cls, pred_resp, pred_bb, label_cls, label_resp, label_bb, ws);
  hipLaunchKernelGGL(yolo_final_kernel, dim3(1), dim3(BLK2), 0, stream, ws, out);
}